// TorchCombineModule_45217415693207
// MI455X (gfx1250) — compile-verified
//
#include <hip/hip_runtime.h>
#include <hip/hip_bf16.h>

// MoE combine: scatter fp32 rows (converted bf16, RNE) per metadata routing.
// Pure bandwidth kernel (~0.77 GB traffic, ~33us floor at 23.3 TB/s).
// CDNA5 path: async global->LDS B128 loads (ASYNCcnt) triple-buffer rows with
// prefetch distance 2; saddr (GVS) addressing; NT cache policy (every byte
// touched exactly once); metadata via scalar loads overlapped with the wait.

#define RNK     1
#define CHIPS   4
#define EXPERTS 8
#define TOKS    2048
#define HID     2048
#define SEQL    2048
#define TPK     8
#define NROWS   (CHIPS * EXPERTS * TOKS)   // 65536 source rows (R=1)
#define RPB     8                          // rows per block
#define BLOCK   256                        // 8 wave32s
#define FPT     (HID / BLOCK)              // 8 floats per thread

typedef unsigned uint4v __attribute__((ext_vector_type(4)));  // native vectors
typedef float    flt4v  __attribute__((ext_vector_type(4)));

// ---- fp32 -> packed bf16 (round-to-nearest-even) ---------------------------
#if __has_builtin(__builtin_amdgcn_cvt_pk_bf16_f32)
typedef __bf16 v2bf __attribute__((ext_vector_type(2)));
static __device__ __forceinline__ unsigned pk_bf16(float a, float b) {
    v2bf r = __builtin_amdgcn_cvt_pk_bf16_f32(a, b);
    return __builtin_bit_cast(unsigned, r);
}
#else
static __device__ __forceinline__ unsigned pk_bf16(float a, float b) {
    unsigned ua = __float_as_uint(a);
    unsigned ub = __float_as_uint(b);
    ua += 0x7fffu + ((ua >> 16) & 1u);
    ub += 0x7fffu + ((ub >> 16) & 1u);
    return (ua >> 16) | (ub & 0xffff0000u);
}
#endif

// ---- async global->LDS row staging (GVS/saddr form, contiguous per wave) ---
// Wave w owns row elements [w*256, w*256+256): lane loads 16B at
// w*1024 + lane*16 bytes; second instruction covers +512B. INST_OFFSET is
// added to BOTH the LDS and global addresses per the CDNA5 ISA, so the LDS
// image stays in linear element order. 2 async ops per row per wave.
static __device__ __forceinline__ void async_row_load(unsigned lds_off, unsigned voff,
                                                      const float* row_base) {
    asm volatile(
        "global_load_async_to_lds_b128 %0, %1, %2 th:TH_LOAD_NT\n\t"
        "global_load_async_to_lds_b128 %0, %1, %2 offset:512 th:TH_LOAD_NT"
        :: "v"(lds_off), "v"(voff), "s"(row_base) : "memory");
}

__global__ __launch_bounds__(BLOCK) void moe_combine_bf16_kernel(
    const float* __restrict__ disp,   // (R,C,E,T,H) fp32
    const int*   __restrict__ meta,   // (R,C,E,T,3) int32
    const int*   __restrict__ cnt,    // (R,C,E)     int32
    __hip_bfloat16* __restrict__ out) // (C,S,K,H)   bf16
{
    __shared__ float buf[3][HID];     // 24 KB triple buffer (prefetch depth 2)
    const int tid  = threadIdx.x;
    const int lane = tid & 31;
    const int wv   = tid >> 5;
    const long long row0 = (long long)blockIdx.x * RPB;

    // Loop-invariant per-lane byte offset into a row (contiguous per wave).
    const unsigned voff = (unsigned)(wv * 1024 + lane * 16);
    // LDS byte offsets (generic pointer low 32 bits == LDS offset on gfx1250).
    const unsigned ldsb[3] = {
        (unsigned)(unsigned long long)&buf[0][0] + voff,
        (unsigned)(unsigned long long)&buf[1][0] + voff,
        (unsigned)(unsigned long long)&buf[2][0] + voff,
    };

    // Prologue: stage rows 0 and 1.
    async_row_load(ldsb[0], voff, disp + row0 * HID);
    async_row_load(ldsb[1], voff, disp + (row0 + 1) * HID);

#pragma unroll
    for (int i = 0; i < RPB; ++i) {
        const long long row = row0 + i;

        // Keep prefetch distance 2: issue row i+2 before consuming row i.
        if (i + 2 < RPB)
            async_row_load(ldsb[(i + 2) % 3], voff, disp + (row + 2) * HID);

        // Metadata scalar loads issued BEFORE the async wait so SMEM latency
        // overlaps it (pinned between the two asm memory barriers).
        const int m0 = meta[3 * row + 0];
        const int m1 = meta[3 * row + 1];
        const int m2 = meta[3 * row + 2];
        const int cv = cnt[(int)(row >> 11)];         // counter for row/T

        // Async loads complete in order: leave only the prefetched rows
        // outstanding -> row i is resident in LDS.
        if (i + 2 < RPB)      asm volatile("s_wait_asynccnt 4" ::: "memory");
        else if (i + 1 < RPB) asm volatile("s_wait_asynccnt 2" ::: "memory");
        else                  asm volatile("s_wait_asynccnt 0" ::: "memory");

        const long long slot = row & (TOKS - 1);      // row % T
        const bool valid = (slot < (long long)cv) && ((m0 % RNK) == 0);

        if (valid) {
            const long long drow = (long long)(m0 / RNK) * (SEQL * TPK)
                                 + (long long)m1 * TPK + (long long)m2;

            const flt4v* lp = (const flt4v*)&buf[i % 3][tid * FPT];
            flt4v va = lp[0];
            flt4v vb = lp[1];

            uint4v u;
            u.x = pk_bf16(va.x, va.y);
            u.y = pk_bf16(va.z, va.w);
            u.z = pk_bf16(vb.x, vb.y);
            u.w = pk_bf16(vb.z, vb.w);

            // 16B-aligned scattered store (4 KB contiguous per row); streamed
            // once -> non-temporal.
            uint4v* dp = (uint4v*)((unsigned short*)out + drow * (long long)HID
                                   + (long long)tid * FPT);
            __builtin_nontemporal_store(u, dp);
        }
    }
}

extern "C" void kernel_launch(void* const* d_in, const int* in_sizes, int n_in,
                              void* d_out, int out_size, void* d_ws, size_t ws_size,
                              hipStream_t stream) {
    (void)in_sizes; (void)n_in; (void)out_size; (void)d_ws; (void)ws_size;
    const float* disp = (const float*)d_in[0];
    const int*   meta = (const int*)d_in[1];
    const int*   cnt  = (const int*)d_in[2];
    __hip_bfloat16* out = (__hip_bfloat16*)d_out;

    dim3 grid(NROWS / RPB);   // 8192 blocks
    dim3 block(BLOCK);        // 256 threads = 8 wave32s
    hipLaunchKernelGGL(moe_combine_bf16_kernel, grid, block, 0, stream,
                       disp, meta, cnt, out);
}